// GatedMemoryModule_6184752906565
// MI455X (gfx1250) — compile-verified
//
#include <hip/hip_runtime.h>

// Gated memory module for MI455X (gfx1250, wave32).
//   simT = [memory; gate_w] @ x^T   (V_WMMA_F32_16X16X4_F32; gate folded in as
//                                    an extra 16-row A-tile, row 64 = gate_w)
//   attn = softmax over 64 slots    (in-lane reduction + one shfl_xor(16))
//   read = attn @ memory            (WMMA, attn re-laid out purely in registers)
//   out  = g*read + (1-g)*x         (blended in WMMA C-layout; coalesced b32 IO)
// Memory-bound: ~1.0 GB HBM traffic => ~44 us floor @ 23.3 TB/s. To actually
// reach that, each wave must keep many global loads in flight: GEMM1's x
// fragments are software-pipelined in groups of 8 (prefetch next group while
// computing current), and the blend x re-loads are issued before the GEMM2
// wmma chain so their latency is covered.

namespace {

constexpr int D = 128;   // MEM_DIM
constexpr int M = 64;    // MEM_SIZE

typedef float v2f __attribute__((ext_vector_type(2)));
typedef float v8f __attribute__((ext_vector_type(8)));

__device__ __forceinline__ v8f wmma4(v2f a, v2f b, v8f c) {
  // D = A(16x4,f32) * B(4x16,f32) + C(16x16,f32)
  return __builtin_amdgcn_wmma_f32_16x16x4_f32(false, a, false, b, (short)0, c,
                                               false, false);
}

// XOR swizzle (flips only addr bits >=2): b64 pairs stay adjacent/aligned,
// row-strided b64 fragment loads are bank-conflict free on 64 banks.
__device__ __forceinline__ int swz128(int row, int col) {
  return row * 128 + (col ^ ((row & 15) << 2));
}

} // namespace

__global__ __launch_bounds__(256, 4)
void gated_memory_wmma(const float* __restrict__ x,
                       const float* __restrict__ mem,
                       const float* __restrict__ gw,
                       const float* __restrict__ gb,
                       float* __restrict__ out,
                       long long nTiles, long long nRows)
{
  // rows 0..63: memory matrix; row 64: gate_w; rows 65..79: zeros  (40 KB)
  __shared__ __align__(16) float memA[80 * 128];
  __shared__ __align__(16) float tailbuf[16 * 128];   // scalar-tail scratch (8 KB)

  const int tid = threadIdx.x;

  for (int i = tid; i < 80 * 128; i += 256) {
    const int row = i >> 7, col = i & 127;
    float v;
    if (row < M)        v = mem[i];
    else if (row == M)  v = gw[col];
    else                v = 0.f;
    memA[swz128(row, col)] = v;
  }
  __syncthreads();

  const int lane = tid & 31;
  const int wave = tid >> 5;
  const int n    = lane & 15;   // fragment column / x-row index within tile
  const int kh   = lane >> 4;   // K-half (0 or 1) of the WMMA fragment
  const float gbias = gb[0];

  for (long long tile = (long long)blockIdx.x * 8 + wave; tile < nTiles;
       tile += (long long)gridDim.x * 8) {
    const long long rowBase = tile * 16;

    // ---- GEMM1: simT(80x16) = [memory; gate_w] @ x^T(128x16) ----
    // acc[t][r] = simT[m = 16t + r + 8*kh][n];  acc[4] tile carries the gate.
    // K processed in 4 groups of 8 chunks; next group's x fragments prefetched
    // while the current group computes -> 8-16 global loads in flight.
    const float* xr = x + (rowBase + n) * D + 2 * kh;
    v8f acc[5];
#pragma unroll
    for (int t = 0; t < 5; ++t)
      acc[t] = (v8f){0.f, 0.f, 0.f, 0.f, 0.f, 0.f, 0.f, 0.f};

    v2f xb[8];
#pragma unroll
    for (int j = 0; j < 8; ++j) xb[j] = *(const v2f*)(xr + 4 * j);

#pragma unroll 1
    for (int grp = 0; grp < 4; ++grp) {
      // prefetch next group's fragments (last group re-reads group 0: cheap
      // cache hits, avoids an out-of-bounds tail and a branch)
      const float* xp = xr + (grp < 3 ? 32 * (grp + 1) : 0);
      v2f nxb[8];
#pragma unroll
      for (int j = 0; j < 8; ++j) nxb[j] = *(const v2f*)(xp + 4 * j);

      const int cbase = 8 * grp;
#pragma unroll
      for (int j = 0; j < 8; ++j) {
#pragma unroll
        for (int t = 0; t < 5; ++t) {
          const v2f av = *(const v2f*)(memA +
              swz128(16 * t + n, 4 * (cbase + j) + 2 * kh));
          acc[t] = wmma4(av, xb[j], acc[t]);
        }
      }
#pragma unroll
      for (int j = 0; j < 8; ++j) xb[j] = nxb[j];
    }

    // ---- gate: row 64 of the product == dot(x[row n], gate_w) ----
    const float s4   = __shfl_xor(acc[4][0], 16);
    const float gdot = (kh ? s4 : acc[4][0]) + gbias;
    const float g    = 1.0f / (1.0f + __expf(-gdot));

    // ---- softmax over the 64 memory slots (axis in-lane + partner half) ----
    float mx = -3.402823466e38f;
#pragma unroll
    for (int t = 0; t < 4; ++t)
#pragma unroll
      for (int r = 0; r < 8; ++r) mx = fmaxf(mx, acc[t][r]);
    mx = fmaxf(mx, __shfl_xor(mx, 16));

    float sum = 0.f;
#pragma unroll
    for (int t = 0; t < 4; ++t)
#pragma unroll
      for (int r = 0; r < 8; ++r) {
        const float e = __expf(acc[t][r] - mx);
        acc[t][r] = e;
        sum += e;
      }
    sum += __shfl_xor(sum, 16);
    const float inv = 1.0f / sum;
#pragma unroll
    for (int t = 0; t < 4; ++t)
#pragma unroll
      for (int r = 0; r < 8; ++r) acc[t][r] *= inv;

    // ---- re-shape attn into GEMM2 A-fragments purely in registers ----
    // A chunk c needs attn[n][m0..m0+1], m0 = 4c + 2*kh. Each partner value is
    // consumed exactly once -> shuffle inline, no materialized swap array.
    v2f areg[16];
#pragma unroll
    for (int c = 0; c < 16; ++c) {
      const int t  = c >> 2;
      const int h  = (c >> 1) & 1;  // which 8-half of the 16-block holds m0
      const int rb = 4 * (c & 1);
      if (h == 0) {
        const float s0 = __shfl_xor(acc[t][rb + 2], 16);
        const float s1 = __shfl_xor(acc[t][rb + 3], 16);
        areg[c][0] = kh ? s0 : acc[t][rb + 0];
        areg[c][1] = kh ? s1 : acc[t][rb + 1];
      } else {
        const float s0 = __shfl_xor(acc[t][rb + 0], 16);
        const float s1 = __shfl_xor(acc[t][rb + 1], 16);
        areg[c][0] = kh ? acc[t][rb + 2] : s0;
        areg[c][1] = kh ? acc[t][rb + 3] : s1;
      }
    }

    // ---- per-output-row gate values (C-layout rows are r + 8*kh) ----
    float garr[8], omg[8];
#pragma unroll
    for (int r = 0; r < 8; ++r) {
      garr[r] = __shfl(g, r + 8 * kh);
      omg[r]  = 1.0f - garr[r];
    }

    // ---- GEMM2: read = attn @ memory, blended & stored per 16-col d-tile ----
    // C-layout: rdv[r] = read[row = r + 8*kh][d = 16*dt + n]; x re-loads are
    // issued BEFORE the wmma chain so the chain hides their (L0/L2) latency;
    // per-row b32 accesses are fully coalesced 64B segments.
    const float* xc = x   + (rowBase + 8 * kh) * D + n;
    float*       oc = out + (rowBase + 8 * kh) * D + n;
#pragma unroll 1
    for (int dt = 0; dt < 8; ++dt) {
      const int dcol = 16 * dt;
      float xv[8];
#pragma unroll
      for (int r = 0; r < 8; ++r) xv[r] = xc[r * D + dcol];

      v8f rdv = {0.f, 0.f, 0.f, 0.f, 0.f, 0.f, 0.f, 0.f};
#pragma unroll
      for (int c = 0; c < 16; ++c) {
        const int m0 = 4 * c + 2 * kh;
        v2f bv;
        bv[0] = memA[swz128(m0,     16 * dt + n)];
        bv[1] = memA[swz128(m0 + 1, 16 * dt + n)];
        rdv = wmma4(areg[c], bv, rdv);
      }
#pragma unroll
      for (int r = 0; r < 8; ++r)
        oc[r * D + dcol] = garr[r] * rdv[r] + omg[r] * xv[r];
    }
  }

  // ---- scalar tail for nRows % 16 (not taken for N = 1e6) ----
  const long long tailBase = nTiles * 16;
  if (blockIdx.x == 0 && tid < 16 && tailBase + tid < nRows) {
    const long long row = tailBase + tid;
    const float* xp = x + row * D;
    float* racc = tailbuf + tid * D;
    for (int d = 0; d < D; ++d) racc[d] = 0.f;
    float mx = -3.402823466e38f;
    for (int m = 0; m < M; ++m) {
      float s = 0.f;
      for (int d = 0; d < D; ++d) s += xp[d] * memA[swz128(m, d)];
      mx = fmaxf(mx, s);
    }
    float sum = 0.f;
    for (int m = 0; m < M; ++m) {
      float s = 0.f;
      for (int d = 0; d < D; ++d) s += xp[d] * memA[swz128(m, d)];
      const float e = __expf(s - mx);
      sum += e;
      for (int d = 0; d < D; ++d) racc[d] += e * memA[swz128(m, d)];
    }
    const float inv = 1.0f / sum;
    float gd = gbias;
    for (int d = 0; d < D; ++d) gd += xp[d] * memA[swz128(M, d)];  // gate_w row
    const float gg = 1.0f / (1.0f + __expf(-gd));
    for (int d = 0; d < D; ++d)
      out[row * D + d] = gg * racc[d] * inv + (1.0f - gg) * xp[d];
  }
}

extern "C" void kernel_launch(void* const* d_in, const int* in_sizes, int n_in,
                              void* d_out, int out_size, void* d_ws, size_t ws_size,
                              hipStream_t stream) {
  (void)n_in; (void)out_size; (void)d_ws; (void)ws_size;
  const float* x   = (const float*)d_in[0];
  const float* mem = (const float*)d_in[1];
  const float* gw  = (const float*)d_in[2];
  const float* gb  = (const float*)d_in[3];
  float* out = (float*)d_out;

  const long long nRows  = (long long)in_sizes[0] / D;  // 1,000,000
  const long long nTiles = nRows / 16;                  // 62,500

  long long want = (nTiles + 7) / 8;      // 8 waves (tiles) per 256-thread block
  if (want < 1) want = 1;
  if (want > 4096) want = 4096;           // grid-stride beyond this
  gated_memory_wmma<<<(int)want, 256, 0, stream>>>(x, mem, gw, gb, out,
                                                   nTiles, nRows);
}